// YOLOXDarkSSL_89593017795398
// MI455X (gfx1250) — compile-verified
//
#include <hip/hip_runtime.h>
#include <hip/hip_bf16.h>
#include <math.h>

typedef __attribute__((ext_vector_type(2))) float v2f;
typedef __attribute__((ext_vector_type(8))) float v8f;

#define Bc 8
#define Nc 64
#define Mc 128
#define Cc 256
#define MAPc 128
#define EPSf 1e-12f
#define MARGINf 0.3f
#define EXTRAf 0.6f

// ---------------------------------------------------------------------------
// 1) Row normalization: dst[r][c] = src[r][c] / sqrt(max(sum_c src^2, EPS))
//    one 256-thread block per row (C == 256), LDS tree reduce (deterministic)
// ---------------------------------------------------------------------------
__global__ void norm_rows_kernel(const float* __restrict__ src,
                                 float* __restrict__ dst)
{
    const int row = blockIdx.x;
    const int c   = threadIdx.x;
    __shared__ float red[Cc];
    float v = src[(size_t)row * Cc + c];
    red[c] = v * v;
    __syncthreads();
    for (int s = Cc / 2; s > 0; s >>= 1) {
        if (c < s) red[c] += red[c + s];
        __syncthreads();
    }
    dst[(size_t)row * Cc + c] = v / sqrtf(fmaxf(red[0], EPSf));
}

// ---------------------------------------------------------------------------
// 2) Feature-map gather + normalize + scale terms.
//    block (b,i) of 256 threads; writes sc_terms[b*64+i] = (1-qn.fv)+(1-kn.fv)
// ---------------------------------------------------------------------------
__global__ void scale_kernel(const float* __restrict__ fm,      // B,C,MAP,MAP
                             const float* __restrict__ targets, // B,N,5
                             const int*   __restrict__ net_h,
                             const int*   __restrict__ net_w,
                             const float* __restrict__ qn,      // B,N,C
                             const float* __restrict__ kn,      // B,N,C
                             float* __restrict__ sc_terms)      // B*N
{
    const int b = blockIdx.x / Nc;
    const int i = blockIdx.x % Nc;
    const int c = threadIdx.x;

    const float cx = targets[((size_t)b * Nc + i) * 5 + 1];
    const float cy = targets[((size_t)b * Nc + i) * 5 + 2];
    const float nw = (float)net_w[0];
    const float nh = (float)net_h[0];
    int ix = (int)floorf(cx / nw * (float)MAPc + 0.5f);
    int iy = (int)floorf(cy / nh * (float)MAPc + 0.5f);
    ix = ix > (MAPc - 1) ? (MAPc - 1) : ix;
    iy = iy > (MAPc - 1) ? (MAPc - 1) : iy;

    const float f = fm[(((size_t)b * Cc + c) * MAPc + iy) * MAPc + ix];

    __shared__ float red[Cc];
    red[c] = f * f;
    __syncthreads();
    for (int s = Cc / 2; s > 0; s >>= 1) {
        if (c < s) red[c] += red[c + s];
        __syncthreads();
    }
    const float fn = f / sqrtf(fmaxf(red[0], EPSf));

    const float dq = qn[((size_t)b * Nc + i) * Cc + c] * fn;
    const float dk = kn[((size_t)b * Nc + i) * Cc + c] * fn;
    __syncthreads();
    red[c] = dq + dk;
    __syncthreads();
    for (int s = Cc / 2; s > 0; s >>= 1) {
        if (c < s) red[c] += red[c + s];
        __syncthreads();
    }
    if (c == 0) sc_terms[b * Nc + i] = 2.0f - red[0];
}

// ---------------------------------------------------------------------------
// 3) WMMA GEMM: S = qn @ kn^T (B,64,64) and L = qn @ nn^T (B,64,128).
//    One wave32 per 16x16 tile, V_WMMA_F32_16X16X4_F32 over K=256 (64 steps).
//    A-layout: lane = {kpair[4], m[3:0]}, two consecutive K floats per lane.
//    B-layout (B = X^T, row-major X): same addressing against the X row.
//    D-layout: VGPR r -> row r (lanes 0-15) / row r+8 (lanes 16-31), col=lane&15.
// ---------------------------------------------------------------------------
__global__ void __launch_bounds__(32)
gemm_tiles_kernel(const float* __restrict__ qn,   // B,64,256
                  const float* __restrict__ kn,   // B,64,256
                  const float* __restrict__ nn,   // B,128,256
                  float* __restrict__ S,          // B,64,64
                  float* __restrict__ L)          // B,64,128
{
    const int tile = blockIdx.x;          // 0 .. 8*48-1
    const int b    = tile / 48;
    const int t    = tile % 48;

    const float* Abase;
    const float* Bbase;
    float*       out;
    int ldo, tm, tn;
    if (t < 16) {                         // S tiles: 4 x 4
        tm = t >> 2; tn = t & 3;
        Abase = qn + (size_t)b * Nc * Cc;
        Bbase = kn + (size_t)b * Nc * Cc;
        out   = S  + (size_t)b * Nc * Nc;
        ldo   = Nc;
    } else {                              // L tiles: 4 x 8
        const int u = t - 16;
        tm = u >> 3; tn = u & 7;
        Abase = qn + (size_t)b * Nc * Cc;
        Bbase = nn + (size_t)b * Mc * Cc;
        out   = L  + (size_t)b * Nc * Mc;
        ldo   = Mc;
    }

    const int lane = threadIdx.x;         // 0..31, EXEC all ones
    const int mrow = lane & 15;
    const int koff = (lane >> 4) * 2;

    const float* ap = Abase + (size_t)(tm * 16 + mrow) * Cc + koff;
    const float* bp = Bbase + (size_t)(tn * 16 + mrow) * Cc + koff;

    v8f acc = {};
#pragma unroll 8
    for (int k = 0; k < Cc; k += 4) {
        v2f a = *(const v2f*)(ap + k);    // 8-byte aligned
        v2f bb = *(const v2f*)(bp + k);
        acc = __builtin_amdgcn_wmma_f32_16x16x4_f32(
            /*neg_a=*/false, a, /*neg_b=*/false, bb,
            /*c_mod=*/(short)0, acc, /*reuse_a=*/false, /*reuse_b=*/false);
    }

    const int col   = lane & 15;
    const int rbase = (lane >> 4) * 8;
#pragma unroll
    for (int r = 0; r < 8; ++r) {
        out[(size_t)(tm * 16 + rbase + r) * ldo + tn * 16 + col] = acc[r];
    }
}

// ---------------------------------------------------------------------------
// 4) Per-batch reduction: triplet, cyclic, SSL, scale sum.
//    One block of 64 threads per batch; thread i owns row i. Fixed-order
//    tree reductions -> deterministic.
//    batch_out[b*4 + {0,1,2,3}] = {sc, trip, cyc, ssl}
// ---------------------------------------------------------------------------
__device__ __forceinline__ float block_reduce64(float v, float* red, int i)
{
    red[i] = v;
    __syncthreads();
    for (int s = 32; s > 0; s >>= 1) {
        if (i < s) red[i] += red[i + s];
        __syncthreads();
    }
    float r = red[0];
    __syncthreads();
    return r;
}

__global__ void reduce_kernel(const float* __restrict__ S,        // B,64,64
                              const float* __restrict__ L,        // B,64,128
                              const float* __restrict__ sc_terms, // B*64
                              float* __restrict__ batch_out)      // B*4
{
    const int b = blockIdx.x;
    const int i = threadIdx.x;            // 0..63
    const float* Sb = S + (size_t)b * Nc * Nc;
    const float* Lb = L + (size_t)b * Nc * Mc;
    __shared__ float red[Nc];

    const float Sii  = Sb[i * Nc + i];
    const float d_ap = sqrtf(fmaxf(2.0f - 2.0f * Sii, EPSf));

    float trip = 0.0f, cyc = 0.0f;
    for (int j = 0; j < Nc; ++j) {
        const float sij = Sb[i * Nc + j];
        const float sji = Sb[j * Nc + i];
        if (j != i) {
            const float dij = sqrtf(fmaxf(2.0f - 2.0f * sij, EPSf));
            const float dji = sqrtf(fmaxf(2.0f - 2.0f * sji, EPSf));
            trip += fmaxf(d_ap - dij + MARGINf, 0.0f);           // t1
            trip += fmaxf(d_ap - dji + MARGINf, 0.0f);           // t2 (D^T)
            cyc  += fabsf(sij - sji);
        }
    }
    for (int j = 0; j < Mc; ++j) {
        const float dqn = sqrtf(fmaxf(2.0f - 2.0f * Lb[i * Mc + j], EPSf));
        trip += fmaxf(d_ap - dqn + MARGINf + EXTRAf, 0.0f);      // t3
    }

    // ce_neg row i: logits = [Sii, L[i,:]] / T   (T = 0.5 -> *2)
    const float lpos = 2.0f * Sii;
    float m1 = lpos;
    for (int j = 0; j < Mc; ++j) m1 = fmaxf(m1, 2.0f * Lb[i * Mc + j]);
    float s1 = __expf(lpos - m1);
    for (int j = 0; j < Mc; ++j) s1 += __expf(2.0f * Lb[i * Mc + j] - m1);
    const float ce_neg = m1 + logf(s1) - lpos;

    // ce_intra row i: logits = S[i,:] / T, pick diagonal
    float m2 = -3.402823466e+38f;
    for (int j = 0; j < Nc; ++j) m2 = fmaxf(m2, 2.0f * Sb[i * Nc + j]);
    float s2 = 0.0f;
    for (int j = 0; j < Nc; ++j) s2 += __expf(2.0f * Sb[i * Nc + j] - m2);
    const float ce_intra = m2 + logf(s2) - lpos;

    const float trip_sum = block_reduce64(trip, red, i);
    const float cyc_sum  = block_reduce64(cyc,  red, i);
    const float ce_sum   = block_reduce64(ce_neg + ce_intra, red, i);
    const float sc_sum   = block_reduce64(sc_terms[b * Nc + i], red, i);

    if (i == 0) {
        batch_out[b * 4 + 0] = sc_sum;
        batch_out[b * 4 + 1] = trip_sum;
        batch_out[b * 4 + 2] = cyc_sum;
        // ssl_b = (mean(ce_neg) + mean(ce_intra)) / num_gt
        batch_out[b * 4 + 3] = (ce_sum / (float)Nc) / (float)Nc;
    }
}

// ---------------------------------------------------------------------------
// 5) Finalizer: reproduce the reference's iterative accumulator recursion.
// ---------------------------------------------------------------------------
__global__ void final_kernel(const float* __restrict__ batch_out,
                             float* __restrict__ out)
{
    const float tcnt = (float)(2 * Nc * (Nc - 1) + Nc * Mc); // 16256
    const float ccnt = (float)(Nc * (Nc - 1));               // 4032
    float scale_loss = 0.0f, trip_loss = 0.0f, cyc_loss = 0.0f, ssl_loss = 0.0f;
    for (int b = 0; b < Bc; ++b) {
        scale_loss = (scale_loss + batch_out[b * 4 + 0]) / (float)Bc;
        trip_loss  = (trip_loss  + batch_out[b * 4 + 1]) / tcnt;
        cyc_loss   = (cyc_loss   + batch_out[b * 4 + 2]) / ccnt;
        ssl_loss   =  ssl_loss   + batch_out[b * 4 + 3];
    }
    const float total = scale_loss + trip_loss + cyc_loss + ssl_loss;
    out[0] = total;
    out[1] = ssl_loss;
    out[2] = cyc_loss;
    out[3] = scale_loss;
    out[4] = trip_loss;
}

// ---------------------------------------------------------------------------
extern "C" void kernel_launch(void* const* d_in, const int* in_sizes, int n_in,
                              void* d_out, int out_size, void* d_ws, size_t ws_size,
                              hipStream_t stream)
{
    const float* q   = (const float*)d_in[0];   // B,N,C
    const float* k   = (const float*)d_in[1];   // B,N,C
    const float* n   = (const float*)d_in[2];   // B,M,C
    const float* fm  = (const float*)d_in[3];   // B,C,MAP,MAP
    const float* tg  = (const float*)d_in[4];   // B,N,5
    const int*   nh  = (const int*)d_in[5];
    const int*   nw  = (const int*)d_in[6];
    float* out = (float*)d_out;

    float* ws = (float*)d_ws;
    float* qn        = ws;                                  // 8*64*256   = 131072
    float* kn        = qn + (size_t)Bc * Nc * Cc;           // 131072
    float* nn        = kn + (size_t)Bc * Nc * Cc;           // 8*128*256  = 262144
    float* Smat      = nn + (size_t)Bc * Mc * Cc;           // 8*64*64    = 32768
    float* Lmat      = Smat + (size_t)Bc * Nc * Nc;         // 8*64*128   = 65536
    float* sc_terms  = Lmat + (size_t)Bc * Nc * Mc;         // 512
    float* batch_out = sc_terms + (size_t)Bc * Nc;          // 32

    norm_rows_kernel<<<Bc * Nc, Cc, 0, stream>>>(q, qn);
    norm_rows_kernel<<<Bc * Nc, Cc, 0, stream>>>(k, kn);
    norm_rows_kernel<<<Bc * Mc, Cc, 0, stream>>>(n, nn);

    scale_kernel<<<Bc * Nc, Cc, 0, stream>>>(fm, tg, nh, nw, qn, kn, sc_terms);

    gemm_tiles_kernel<<<Bc * 48, 32, 0, stream>>>(qn, kn, nn, Smat, Lmat);

    reduce_kernel<<<Bc, Nc, 0, stream>>>(Smat, Lmat, sc_terms, batch_out);

    final_kernel<<<1, 1, 0, stream>>>(batch_out, out);
}